// noCluster_55568286876312
// MI455X (gfx1250) — compile-verified
//
#include <hip/hip_runtime.h>
#include <hip/hip_bf16.h>
#include <math.h>

typedef __attribute__((ext_vector_type(2))) float v2f;
typedef __attribute__((ext_vector_type(8))) float v8f;

// ---------------------------------------------------------------------------
// Kernel 1: EmbeddingBag(mean) -- one block (64 threads = D/4 float4 lanes)
// per mention. Token index t is block-uniform -> feature_seq[t] is a scalar
// load; each embedding row is 64 coalesced global_load_b128. The next row is
// prefetched (global_prefetch_b8) while the current one is summed to break
// the serial row-latency chain. This kernel carries ~512MB (the roofline).
// ---------------------------------------------------------------------------
__global__ __launch_bounds__(64) void mention_mean_kernel(
    const int* __restrict__ feature_seq, const int* __restrict__ offset_seq,
    const float4* __restrict__ we4, float4* __restrict__ men4,
    int M, int T, int D4) {
  int m = blockIdx.x;
  int d = threadIdx.x;                    // 0..D4-1
  int s = offset_seq[m];
  int e = (m == M - 1) ? T : offset_seq[m + 1];
  float4 acc = make_float4(0.f, 0.f, 0.f, 0.f);
  int f = feature_seq[s];
  for (int t = s; t < e; ++t) {
    int fn = (t + 1 < e) ? feature_seq[t + 1] : f;
    __builtin_prefetch(we4 + (size_t)fn * D4 + d, 0, 1);   // global_prefetch_b8
    float4 v = we4[(size_t)f * D4 + d];
    acc.x += v.x; acc.y += v.y; acc.z += v.z; acc.w += v.w;
    f = fn;
  }
  float inv = 1.0f / (float)(e - s);
  acc.x *= inv; acc.y *= inv; acc.z *= inv; acc.w *= inv;
  men4[(size_t)m * D4 + d] = acc;
}

// ---------------------------------------------------------------------------
// Kernel 2: type_idx[b] = argmax_k typeTensor[b, k]  (first-max, like jnp)
// ---------------------------------------------------------------------------
__global__ void type_argmax_kernel(const float* __restrict__ typeTensor,
                                   int* __restrict__ type_idx, int B, int K) {
  int b = blockIdx.x * blockDim.x + threadIdx.x;
  if (b >= B) return;
  const float* row = typeTensor + (size_t)b * K;
  float best = row[0];
  int bi = 0;
  for (int k = 1; k < K; ++k) {
    float v = row[k];
    if (v > best) { best = v; bi = k; }
  }
  type_idx[b] = bi;
}

// ---------------------------------------------------------------------------
// Kernel 3: bag_id[m] = b for m in [scope[b], scope[b+1])
// ---------------------------------------------------------------------------
__global__ void bag_assign_kernel(const int* __restrict__ scope,
                                  int* __restrict__ bag_id, int B) {
  int b = blockIdx.x * blockDim.x + threadIdx.x;
  if (b >= B) return;
  int s = scope[b], e = scope[b + 1];
  for (int m = s; m < e; ++m) bag_id[m] = b;
}

// ---------------------------------------------------------------------------
// Kernel 4: sel[m] = dot(men_emb[m,:], W[type_idx[bag_id[m]],:])
// One wave32 per mention; float4 loads (2 x b128 per lane per array),
// then wave32 shuffle reduction. (The reference's full [M,K] GEMM is never
// materialized -- only this one column per mention is ever consumed.)
// ---------------------------------------------------------------------------
__global__ __launch_bounds__(256) void sel_kernel(
    const float* __restrict__ men_emb, const float* __restrict__ W,
    const int* __restrict__ bag_id, const int* __restrict__ type_idx,
    float* __restrict__ sel, int M, int D) {
  int gtid = blockIdx.x * blockDim.x + threadIdx.x;
  int m = gtid >> 5;          // wave32
  int lane = threadIdx.x & 31;
  if (m >= M) return;
  int ti = type_idx[bag_id[m]];
  const float4* a4 = (const float4*)(men_emb + (size_t)m * D);
  const float4* w4 = (const float4*)(W + (size_t)ti * D);
  int D4 = D >> 2;
  float p = 0.0f;
  for (int j = lane; j < D4; j += 32) {
    float4 av = a4[j];
    float4 wv = w4[j];
    p += av.x * wv.x + av.y * wv.y + av.z * wv.z + av.w * wv.w;
  }
  for (int off = 16; off > 0; off >>= 1) p += __shfl_down(p, off, 32);
  if (lane == 0) sel[m] = p;
}

// ---------------------------------------------------------------------------
// Kernel 5: per-bag stable softmax over sel + attention-weighted sum of
// men_emb -> bag_emb[b,:]. One block (64 threads = D/4) per bag; bags
// average ~10 mentions so the thread-0 serial max/sum is cheap; the per-lane
// exp recompute is a handful of v_exp ops against 1KB row reads.
// ---------------------------------------------------------------------------
__global__ __launch_bounds__(64) void bag_reduce_kernel(
    const float4* __restrict__ men4, const float* __restrict__ sel,
    const int* __restrict__ scope, float4* __restrict__ bag4, int D4) {
  int b = blockIdx.x;
  __shared__ float s_mx, s_sum;
  int s = scope[b], e = scope[b + 1];
  if (threadIdx.x == 0) {
    float mx = -INFINITY;
    for (int m = s; m < e; ++m) mx = fmaxf(mx, sel[m]);
    float sum = 0.0f;
    for (int m = s; m < e; ++m) sum += __expf(sel[m] - mx);
    s_mx = mx;
    s_sum = sum;
  }
  __syncthreads();
  float mx = s_mx;
  float inv = 1.0f / s_sum;
  int d = threadIdx.x;
  float4 acc = make_float4(0.f, 0.f, 0.f, 0.f);
  for (int m = s; m < e; ++m) {
    float att = __expf(sel[m] - mx) * inv;
    float4 v = men4[(size_t)m * D4 + d];
    acc.x += att * v.x; acc.y += att * v.y;
    acc.z += att * v.z; acc.w += att * v.w;
  }
  bag4[(size_t)b * D4 + d] = acc;
}

// ---------------------------------------------------------------------------
// Kernel 6: out[B,K] = bag_emb[B,D] @ W[K,D]^T using V_WMMA_F32_16X16X4_F32
// (fp32-exact, matches the fp32 reference). One wave32 per 16x16 output
// tile, 64 K-steps of 4 (D=256). ISA f32 16x4 A layout: lanes 0-15 hold
// K+{0,1} in v0/v1, lanes 16-31 hold K+{2,3}; B (4x16) mirrors the C/D
// half-split. D layout: VGPR r -> row tm*16+r+8*(lane>=16), col tn*16+l15.
// Edge tiles clamp load rows and mask stores against B=5000 / K=100.
// ---------------------------------------------------------------------------
__global__ __launch_bounds__(32) void bag_out_gemm_kernel(
    const float* __restrict__ bag_emb, const float* __restrict__ W,
    float* __restrict__ out, int B, int K, int D, int tilesN) {
  int tm = blockIdx.x / tilesN;
  int tn = blockIdx.x % tilesN;
  int lane = threadIdx.x;
  int half = lane >> 4;       // 0: lanes 0-15, 1: lanes 16-31
  int l15 = lane & 15;

  int ra = tm * 16 + l15; if (ra >= B) ra = B - 1;   // clamped A row
  int cb = tn * 16 + l15; if (cb >= K) cb = K - 1;   // clamped W row (out col)
  const float* Arow = bag_emb + (size_t)ra * D;
  const float* Brow = W + (size_t)cb * D;

  v8f c = {};
  for (int k0 = 0; k0 < D; k0 += 4) {
    int ka = k0 + half * 2;
    v2f a, bvec;
    a.x = Arow[ka];     a.y = Arow[ka + 1];
    bvec.x = Brow[ka];  bvec.y = Brow[ka + 1];
    // 8 args: (neg_a, A, neg_b, B, c_mod, C, reuse_a, reuse_b)
    c = __builtin_amdgcn_wmma_f32_16x16x4_f32(
        false, a, false, bvec, (short)0, c, false, false);
  }

  int col = tn * 16 + l15;
  if (col < K) {
    for (int r = 0; r < 8; ++r) {
      int row = tm * 16 + r + half * 8;
      if (row < B) out[(size_t)row * K + col] = c[r];
    }
  }
}

// ---------------------------------------------------------------------------
extern "C" void kernel_launch(void* const* d_in, const int* in_sizes, int n_in,
                              void* d_out, int out_size, void* d_ws, size_t ws_size,
                              hipStream_t stream) {
  const int*   feature_seq    = (const int*)d_in[0];   // [T]
  const int*   offset_seq     = (const int*)d_in[1];   // [M]
  const int*   scope          = (const int*)d_in[2];   // [B+1]
  const float* typeTensor     = (const float*)d_in[3]; // [B,K]
  const float* word_embedding = (const float*)d_in[4]; // [V,D]
  const float* linear_weight  = (const float*)d_in[5]; // [K,D]
  float* out = (float*)d_out;

  const int T = in_sizes[0];
  const int M = in_sizes[1];
  const int B = in_sizes[2] - 1;
  const int K = in_sizes[3] / B;
  const int D = in_sizes[5] / K;
  const int D4 = D >> 2;

  // workspace layout (bytes): men_emb | bag_emb | sel | bag_id | type_idx
  char* ws = (char*)d_ws;
  float* men_emb = (float*)ws;  ws += (size_t)M * D * sizeof(float);
  float* bag_emb = (float*)ws;  ws += (size_t)B * D * sizeof(float);
  float* sel     = (float*)ws;  ws += (size_t)M * sizeof(float);
  int*   bag_id  = (int*)ws;    ws += (size_t)M * sizeof(int);
  int*   type_idx= (int*)ws;    ws += (size_t)B * sizeof(int);

  mention_mean_kernel<<<M, D4, 0, stream>>>(
      feature_seq, offset_seq, (const float4*)word_embedding,
      (float4*)men_emb, M, T, D4);

  type_argmax_kernel<<<(B + 255) / 256, 256, 0, stream>>>(
      typeTensor, type_idx, B, K);

  bag_assign_kernel<<<(B + 255) / 256, 256, 0, stream>>>(scope, bag_id, B);

  sel_kernel<<<(M * 32 + 255) / 256, 256, 0, stream>>>(
      men_emb, linear_weight, bag_id, type_idx, sel, M, D);

  bag_reduce_kernel<<<B, D4, 0, stream>>>(
      (const float4*)men_emb, sel, scope, (float4*)bag_emb, D4);

  const int tilesM = (B + 15) / 16;
  const int tilesN = (K + 15) / 16;
  bag_out_gemm_kernel<<<tilesM * tilesN, 32, 0, stream>>>(
      bag_emb, linear_weight, out, B, K, D, tilesN);
}